// CandidateFinder_14877766713592
// MI455X (gfx1250) — compile-verified
//
#include <hip/hip_runtime.h>
#include <hip/hip_bf16.h>

typedef __attribute__((ext_vector_type(2))) float v2f;
typedef __attribute__((ext_vector_type(8))) float v8f;

#define LSEQ   2048
#define DDIM   64
#define NHASH  8
#define KMAX   64
#define NBUCK  256

// ---------------------------------------------------------------------------
// Kernel 1: hash codes via V_WMMA_F32_16X16X4_F32 (exact f32 accumulation).
// One wave computes codes for 16 rows. A = binarized x tile (16x4 per step),
// B = W tile (4x16, cols 8..15 zero), C accumulates the 16x16 projection.
// A layout (ISA 7.12.2): lanes 0-15 hold M=0..15; VGPR0 = K0 (lanes<16) / K2
// (lanes>=16), VGPR1 = K1 / K3.  B: VGPRv holds K=v (lanes<16), K=v+2 (>=16),
// N = lane&15.  D: VGPRr holds M=r (lanes<16) / M=r+8 (>=16), N = lane&15.
// All operand construction is branchless (selects, not guarded loads) so the
// wave keeps EXEC = all-1s through the whole WMMA loop with no save/restore.
// ---------------------------------------------------------------------------
__global__ void __launch_bounds__(128)
lsh_hash_codes_wmma(const float* __restrict__ x, const float* __restrict__ W,
                    int* __restrict__ code, int nrows) {
    const int lane = threadIdx.x & 31;
    const int wave = threadIdx.x >> 5;
    const int tile = blockIdx.x * (blockDim.x >> 5) + wave;   // 16 rows / tile
    const int row0 = tile * 16;
    if (row0 >= nrows) return;               // wave-uniform: EXEC stays all-1s

    const int half = lane >> 4;              // 0: lanes 0-15, 1: lanes 16-31
    const int lid  = lane & 15;
    const int col  = lid & (NHASH - 1);      // clamped W column (always valid)
    const float colmask = (lid < NHASH) ? 1.0f : 0.0f;

    const float* __restrict__ arow = x + (size_t)(row0 + lid) * DDIM;

    v8f c = {0.f, 0.f, 0.f, 0.f, 0.f, 0.f, 0.f, 0.f};

#pragma unroll
    for (int k0 = 0; k0 < DDIM; k0 += 4) {
        const int kk = k0 + half * 2;
        // A: binarized query/key tile (exactly 0.0 / 1.0 -> exact in f32)
        v2f a;
        a.x = (arow[kk + 0] > 0.f) ? 1.f : 0.f;
        a.y = (arow[kk + 1] > 0.f) ? 1.f : 0.f;
        // B: W is [64][8] row-major; unconditional loads, lanes 8..15 zeroed
        // via multiply-select (no exec divergence in the WMMA loop).
        v2f b;
        b.x = W[(kk + 0) * NHASH + col] * colmask;
        b.y = W[(kk + 1) * NHASH + col] * colmask;
        // 8 args: (neg_a, A, neg_b, B, c_mod, C, reuse_a, reuse_b)
        c = __builtin_amdgcn_wmma_f32_16x16x4_f32(
                false, a, false, b, (short)0, c, false, false);
    }

    // Pack sign bits: row m = r + 8*half sits in accumulator r; hash bit n of
    // row r is lane n (n<8), of row r+8 is lane 16+n.  One ballot per r gives
    // both rows' codes.
#pragma unroll
    for (int r = 0; r < 8; ++r) {
        const unsigned m = __builtin_amdgcn_ballot_w32(c[r] > 0.0f);
        if (lane == r) {
            code[row0 + r] = (int)(m & 0xFFu);
        } else if (lane == r + 8) {
            code[row0 + r + 8] = (int)((m >> 16) & 0xFFu);
        }
    }
}

// ---------------------------------------------------------------------------
// Kernel 2: per (batch, bucket) gather of the first KMAX key indices whose
// code matches the bucket, padded with -1.  4 blocks x 256 threads; all
// threads of a block stream the same 8KB of codes (broadcast, L2-resident).
// ---------------------------------------------------------------------------
__global__ void __launch_bounds__(NBUCK)
lsh_build_buckets(const int* __restrict__ kcode, int* __restrict__ bucket) {
    const int b    = blockIdx.x;
    const int want = threadIdx.x;                       // bucket id 0..255
    const int* __restrict__ kc = kcode + b * LSEQ;
    int* __restrict__ out = bucket + ((size_t)b * NBUCK + want) * KMAX;

    int cnt = 0;
    for (int i = 0; i < LSEQ && cnt < KMAX; ++i) {
        if (kc[i] == want) out[cnt++] = i;
    }
    for (; cnt < KMAX; ++cnt) out[cnt] = -1;
}

// ---------------------------------------------------------------------------
// Kernel 3: blind copy bucket list -> output (int64).  One thread per output
// element; bucket reads and int64 writes are fully coalesced.
// ---------------------------------------------------------------------------
__global__ void __launch_bounds__(256)
lsh_emit(const int* __restrict__ qcode, const int* __restrict__ bucket,
         long long* __restrict__ out) {
    const int gid = blockIdx.x * blockDim.x + threadIdx.x; // B*L*KMAX threads
    const int j   = gid & (KMAX - 1);
    const int q   = gid >> 6;                              // b*L + lq
    const int b   = q / LSEQ;
    const int cod = qcode[q];
    out[gid] = (long long)bucket[((size_t)b * NBUCK + cod) * KMAX + j];
}

extern "C" void kernel_launch(void* const* d_in, const int* in_sizes, int n_in,
                              void* d_out, int out_size, void* d_ws, size_t ws_size,
                              hipStream_t stream) {
    const float* query = (const float*)d_in[0];   // [4,2048,64] f32
    const float* key   = (const float*)d_in[1];   // [4,2048,64] f32
    const float* W     = (const float*)d_in[2];   // [64,8] f32
    // d_in[3] = head_idx (unused by the reference computation)

    const int B = 4;
    const int nrows = B * LSEQ;                   // 8192 rows per tensor

    // Workspace layout
    int* qcode  = (int*)d_ws;                                    // 8192 ints
    int* kcode  = qcode + nrows;                                 // 8192 ints
    int* bucket = kcode + nrows;                                 // 4*256*64 ints

    // Hash codes: 512 tiles of 16 rows, 4 waves (128 threads) per block
    const int tiles  = nrows / 16;                // 512
    const int hblk   = 128;                       // 4 wave32s
    const int hgrid  = tiles / (hblk / 32);       // 128 blocks
    lsh_hash_codes_wmma<<<hgrid, hblk, 0, stream>>>(query, W, qcode, nrows);
    lsh_hash_codes_wmma<<<hgrid, hblk, 0, stream>>>(key,   W, kcode, nrows);

    // Bucket build: one thread per (batch, bucket)
    lsh_build_buckets<<<B, NBUCK, 0, stream>>>(kcode, bucket);

    // Emit: one thread per output int64
    const int total = B * LSEQ * KMAX;            // 524288
    lsh_emit<<<total / 256, 256, 0, stream>>>(qcode, bucket, (long long*)d_out);
}